// character_GraphConvolution_37469294690434
// MI455X (gfx1250) — compile-verified
//
#include <hip/hip_runtime.h>
#include <cstdint>

// ---------------------------------------------------------------------------
// COO SpMM:  out[rows[e], :] += vals[e] * input[cols[e], :]   (D = 128, f32)
//
// MI455X reasoning:
//  - 154 MFLOP vs ~58 MB compulsory HBM traffic -> pure memory problem
//    (~2.5 us floor at 23.3 TB/s). input (25.6 MB) + out (25.6 MB) both fit
//    in the 192 MB L2, so the random gather and the 76.8M f32 scatter-atomics
//    are L2-resident after first touch.
//  - WMMA is inapplicable: every edge addresses a random row; there is no
//    dense 16x16 tile structure, and building selection matrices would cost
//    more memory ops than the trivial FLOPs saved.
//  - CDNA5 path used instead: one wave32 per edge; a single
//    global_load_async_to_lds_b128 stages the whole 512 B feature row
//    (16 B/lane x 32 lanes) into a per-wave LDS double buffer, tracked by
//    ASYNCcnt and retired with s_wait_asynccnt (async loads complete in
//    order). Edge metadata is readfirstlane-scalarized so it rides the SMEM/
//    KMcnt path, keeping the vector pipe free for gather + atomics.
// ---------------------------------------------------------------------------

#define D_FEAT 128
#define WAVES_PER_BLOCK 8   // blockDim.x = 256 -> 8 wave32s on CDNA5

__global__ void zero_out_kernel(float4* __restrict__ out, long long n4) {
    long long i = (long long)blockIdx.x * blockDim.x + threadIdx.x;
    long long s = (long long)gridDim.x * blockDim.x;
    for (; i < n4; i += s) {
        out[i] = make_float4(0.f, 0.f, 0.f, 0.f);
    }
}

__global__ __launch_bounds__(256) void spmm_scatter_kernel(
    const float* __restrict__ input,   // [N, 128]
    const int*   __restrict__ rows,    // [E]
    const int*   __restrict__ cols,    // [E]
    const float* __restrict__ vals,    // [E]
    float*       __restrict__ out,     // [N, 128]
    int n_edges)
{
    // Per-wave private double buffer: 2 stages x 128 floats (512 B) per wave.
    __shared__ float buf[WAVES_PER_BLOCK][2][D_FEAT];

    const int lane = threadIdx.x & 31;
    // Wave index is uniform across the wave; readfirstlane moves it (and all
    // edge-metadata address math derived from it) onto the scalar unit, so
    // rows/cols/vals loads become s_load_b32 (KMcnt) instead of vector loads.
    const int wave = __builtin_amdgcn_readfirstlane(threadIdx.x >> 5);

    const long long gwave  = (long long)blockIdx.x * WAVES_PER_BLOCK + wave;
    const long long stride = (long long)gridDim.x * WAVES_PER_BLOCK;

    // LDS byte addresses (low 32 bits of the flat LDS-aperture address carry
    // the LDS offset on CDNA5) for this lane's 16-byte slot in each stage.
    const uint32_t lds0 = (uint32_t)(uintptr_t)&buf[wave][0][lane * 4];
    const uint32_t lds1 = (uint32_t)(uintptr_t)&buf[wave][1][lane * 4];

    long long e = gwave;

    // Prologue: stage the first edge's source row into buffer 0.
    if (e < n_edges) {
        const int c = cols[e];
        const uint64_t ga =
            (uint64_t)(uintptr_t)(input + (size_t)c * D_FEAT + lane * 4);
        asm volatile("global_load_async_to_lds_b128 %0, %1, off"
                     :: "v"(lds0), "v"(ga)
                     : "memory");
    }

    int tog = 0;
    for (; e < n_edges; e += stride) {
        // Scalar-path metadata for the CURRENT edge, issued before the async
        // prefetch/wait so it overlaps the async completion window.
        const int   r = rows[e];
        const float w = vals[e];

        const long long nxt = e + stride;
        if (nxt < n_edges) {
            // Prefetch next edge's row into the other stage, then wait until
            // only that newest async op is outstanding -> current stage ready
            // (async loads complete in order).
            const int c = cols[nxt];
            const uint64_t ga =
                (uint64_t)(uintptr_t)(input + (size_t)c * D_FEAT + lane * 4);
            const uint32_t l = tog ? lds0 : lds1;
            asm volatile("global_load_async_to_lds_b128 %0, %1, off"
                         :: "v"(l), "v"(ga)
                         : "memory");
            asm volatile("s_wait_asynccnt 0x1" ::: "memory");
        } else {
            asm volatile("s_wait_asynccnt 0x0" ::: "memory");
        }

        // ds_load_b128 of the staged row (16 B per lane).
        const float4 v = *(const float4*)&buf[wave][tog][lane * 4];

        float* dst = out + (size_t)r * D_FEAT + lane * 4;
        atomicAdd(dst + 0, v.x * w);
        atomicAdd(dst + 1, v.y * w);
        atomicAdd(dst + 2, v.z * w);
        atomicAdd(dst + 3, v.w * w);

        tog ^= 1;
    }
}

extern "C" void kernel_launch(void* const* d_in, const int* in_sizes, int n_in,
                              void* d_out, int out_size, void* d_ws, size_t ws_size,
                              hipStream_t stream) {
    const float* input = (const float*)d_in[0];   // [N, 128] f32
    const int*   rows  = (const int*)  d_in[1];   // [E] i32
    const int*   cols  = (const int*)  d_in[2];   // [E] i32
    const float* vals  = (const float*)d_in[3];   // [E] f32

    float* out = (float*)d_out;                   // [N, 128] f32
    const int n_edges = in_sizes[1];

    // 1) Zero the output (harness poisons it; segment_sum starts from 0).
    const long long n4 = (long long)out_size / 4;
    {
        const int threads = 256;
        int blocks = (int)((n4 + threads - 1) / threads);
        if (blocks > 4096) blocks = 4096;
        if (blocks < 1)    blocks = 1;
        zero_out_kernel<<<blocks, threads, 0, stream>>>((float4*)out, n4);
    }

    // 2) Edge scatter: 8 wave32s per block, one edge per wave per iteration,
    //    ~37 edges per wave so the async double-buffer pipeline has depth.
    {
        const int threads = 256;                  // 8 waves
        const int blocks  = 2048;                 // 16384 waves total
        spmm_scatter_kernel<<<blocks, threads, 0, stream>>>(
            input, rows, cols, vals, out, n_edges);
    }
}